// cov1d_7456063225857
// MI455X (gfx1250) — compile-verified
//
#include <hip/hip_runtime.h>

// 1-D conv, K=5, stride=2, fp32. Memory-bound (AI ~0.83 flop/byte, ~8.6us floor
// at 23.3 TB/s): stream inputs once via gfx1250 async global->LDS DMA, compute
// from LDS, write non-temporal coalesced b128 stores. Bounds checks hoisted to
// uniform per-block branches so 4095/4096 blocks run the branch-free fast path.

#define KSZ 5
#define STEP 2
#define THREADS 256
#define OUT_PER_BLOCK 4096
#define IN_FLOATS (OUT_PER_BLOCK * STEP + KSZ - 1)   // 8196
#define IN_CHUNKS ((IN_FLOATS + 3) / 4)              // 2049 x 16B chunks

typedef int   vi4 __attribute__((ext_vector_type(4)));
typedef float vf4 __attribute__((ext_vector_type(4)));
typedef __attribute__((address_space(1))) vi4 g_vi4;   // global
typedef __attribute__((address_space(3))) vi4 l_vi4;   // LDS
typedef __attribute__((address_space(3))) void as3_void;

__device__ __forceinline__ void async_copy_b128(const float* gsrc, float* ldst) {
#if __has_builtin(__builtin_amdgcn_global_load_async_to_lds_b128)
    __builtin_amdgcn_global_load_async_to_lds_b128(
        (g_vi4*)gsrc, (l_vi4*)ldst, 0, 0);
#else
    unsigned lds_off = (unsigned)(unsigned long long)(as3_void*)ldst;
    asm volatile("global_load_async_to_lds_b128 %0, %1, off"
                 :: "v"(lds_off), "v"(gsrc) : "memory");
#endif
}

__device__ __forceinline__ void wait_async0() {
#if __has_builtin(__builtin_amdgcn_s_wait_asynccnt)
    __builtin_amdgcn_s_wait_asynccnt(0);
#else
    asm volatile("s_wait_asynccnt 0" ::: "memory");
#endif
}

__global__ __launch_bounds__(THREADS)
void conv1d_k5s2_async(const float* __restrict__ x,
                       const float* __restrict__ w,
                       float* __restrict__ out,
                       long n, long out_n) {
    __shared__ float s[IN_CHUNKS * 4];   // 32784 B of the 320 KB WGP LDS

    const int  tid        = threadIdx.x;
    const long blockStart = (long)blockIdx.x * (OUT_PER_BLOCK * STEP); // input idx
    const float* gbase    = x + blockStart;

    // ---- Stage input tile into LDS with async DMA (ASYNCcnt path) ----
    const bool fullIn = (blockStart + (long)IN_CHUNKS * 4) <= n;  // uniform
    if (fullIn) {
        // branch-free fast path: pure clause of async b128 copies
        for (int c = tid; c < IN_CHUNKS; c += THREADS)
            async_copy_b128(gbase + (c << 2), &s[c << 2]);
    } else {
        // last block only: per-chunk guard + zero-fill past tensor end
        for (int c = tid; c < IN_CHUNKS; c += THREADS) {
            long g = blockStart + (long)c * 4;
            if (g + 4 <= n) {
                async_copy_b128(gbase + (c << 2), &s[c << 2]);
            } else {
                #pragma unroll
                for (int e = 0; e < 4; ++e)
                    s[c * 4 + e] = (g + e < n) ? x[g + e] : 0.0f;
            }
        }
    }
    wait_async0();      // my wave's async transfers are in LDS
    __syncthreads();    // everyone's are

    // ---- Compute: 4 rows x 4 contiguous outputs per thread ----
    const float w0 = w[0], w1 = w[1], w2 = w[2], w3 = w[3], w4 = w[4];
    const long outBase = (long)blockIdx.x * OUT_PER_BLOCK;
    const bool fullOut = (outBase + OUT_PER_BLOCK) <= out_n;      // uniform

    #pragma unroll
    for (int r = 0; r < 4; ++r) {
        const int ol = r * (THREADS * 4) + tid * 4;      // local output index
        const int ib = STEP * ol;                        // local input index (32B aligned)
        const vf4 a = *(const vf4*)&s[ib];
        const vf4 b = *(const vf4*)&s[ib + 4];
        const vf4 c = *(const vf4*)&s[ib + 8];

        vf4 o;
        o.x = w0*a.x + w1*a.y + w2*a.z + w3*a.w + w4*b.x;
        o.y = w0*a.z + w1*a.w + w2*b.x + w3*b.y + w4*b.z;
        o.z = w0*b.x + w1*b.y + w2*b.z + w3*b.w + w4*c.x;
        o.w = w0*b.z + w1*b.w + w2*c.x + w3*c.y + w4*c.z;

        long oi = outBase + ol;
        if (fullOut) {
            // streamed once, never re-read: non-temporal b128 store
            __builtin_nontemporal_store(o, (vf4*)&out[oi]);
        } else {
            if (oi     < out_n) out[oi]     = o.x;
            if (oi + 1 < out_n) out[oi + 1] = o.y;
            if (oi + 2 < out_n) out[oi + 2] = o.z;
            if (oi + 3 < out_n) out[oi + 3] = o.w;
        }
    }
}

extern "C" void kernel_launch(void* const* d_in, const int* in_sizes, int n_in,
                              void* d_out, int out_size, void* d_ws, size_t ws_size,
                              hipStream_t stream) {
    const float* x = (const float*)d_in[0];
    const float* w = (const float*)d_in[1];
    float*     out = (float*)d_out;

    const long n     = (long)in_sizes[0];
    const long out_n = (long)out_size;
    const int  nblk  = (int)((out_n + OUT_PER_BLOCK - 1) / OUT_PER_BLOCK);

    conv1d_k5s2_async<<<dim3(nblk), dim3(THREADS), 0, stream>>>(x, w, out, n, out_n);
}